// DPLSafePolicy_32701880992284
// MI455X (gfx1250) — compile-verified
//
#include <hip/hip_runtime.h>
#include <math.h>

typedef __attribute__((ext_vector_type(2))) float v2f;
typedef __attribute__((ext_vector_type(4))) float v4f;
typedef __attribute__((ext_vector_type(8))) float v8f;

#define WAVES_PER_BLOCK 4
#define BLOCK (WAVES_PER_BLOCK * 32)
#define KCHUNK 512   // K elements staged per LDS chunk (32 KB)

// One wave computes a 16(row) x 16(col) tile of Y = x @ Wcat, where
// Wcat[H,16] = [Wg | Wp | Wa | 0], using V_WMMA_F32_16X16X4_F32 with a
// K-loop of H/4 steps. Epilogue computes the DeepProbLog safe-policy
// posterior per row and writes [B,5].
//
// 4 waves/block -> 256 workgroups at B=16384: spreads the HBM stream over
// more WGPs than the 8-wave/128-block variant while keeping the identical
// per-wave inner loop (clause of 8 global_load_b64 + ds_load_2addr_b64 +
// 8x v_wmma_f32_16x16x4_f32).
__global__ __launch_bounds__(BLOCK) void dpl_safe_policy_kernel(
    const float* __restrict__ x,
    const float* __restrict__ Wg, const float* __restrict__ bg,
    const float* __restrict__ Wp, const float* __restrict__ bp,
    const float* __restrict__ Wa, const float* __restrict__ ba,
    float* __restrict__ out, int H)
{
  // ldsW: B-operand staging, pre-swizzled as pairs: for k-pair p=k/2, col n:
  //   ldsW[(p*16 + n)*2 + (k&1)] = Wcat[k][n]
  // so lane (half h, col n) reads its {K=k+2h, K=k+2h+1} fragment as one b64.
  __shared__ float ldsW[KCHUNK * 16];            // 32 KB
  __shared__ float ldsC[WAVES_PER_BLOCK * 256];  // 4 KB, per-wave C-tile scratch

  const int tid  = threadIdx.x;
  const int lane = tid & 31;
  const int wave = tid >> 5;

  const int tile    = blockIdx.x * WAVES_PER_BLOCK + wave;
  const int rowBase = tile * 16;

  // A-fragment addressing (ISA 32-bit A 16x4 layout):
  //   lane 0-15:  M=lane,    holds K = k+0, k+1
  //   lane 16-31: M=lane-16, holds K = k+2, k+3
  const int m     = lane & 15;
  const int half  = lane >> 4;
  const float* aPtr = x + (size_t)(rowBase + m) * (size_t)H + (half << 1);

  // B-fragment base in LDS: pair index advances with k; lane half adds +1 pair.
  const float* bPtr = ldsW + ((half * 16 + m) << 1);

  v8f c = {};

  for (int k0 = 0; k0 < H; k0 += KCHUNK) {
    __syncthreads();  // previous chunk fully consumed by all waves

    // Cooperative stage of Wcat[k0 .. k0+KCHUNK) into swizzled LDS layout.
    for (int idx = tid; idx < KCHUNK * 16; idx += BLOCK) {
      int kk = idx >> 4;          // 0..KCHUNK-1
      int k  = k0 + kk;
      int n  = idx & 15;
      float v;
      if (n < 4)       v = Wg[k * 4 + n];
      else if (n < 8)  v = Wp[k * 4 + (n - 4)];
      else if (n < 13) v = Wa[k * 5 + (n - 8)];
      else             v = 0.0f;
      ldsW[(((kk >> 1) * 16 + n) << 1) + (kk & 1)] = v;
    }
    __syncthreads();

    // Prefetch next chunk of x for this wave's rows (global_prefetch_b8).
    if (k0 + KCHUNK < H)
      __builtin_prefetch(aPtr + k0 + KCHUNK, 0, 1);

#pragma unroll 8
    for (int k = 0; k < KCHUNK; k += 4) {
      v2f a = *(const v2f*)(aPtr + k0 + k);          // global_load_b64
      v2f b = *(const v2f*)(bPtr + (k >> 1) * 32);   // ds_load_2addr_b64 (paired)
      c = __builtin_amdgcn_wmma_f32_16x16x4_f32(
              /*neg_a=*/false, a, /*neg_b=*/false, b,
              /*c_mod=*/(short)0, c, /*reuse_a=*/false, /*reuse_b=*/false);
    }
  }

  // Spill C tile (ISA C layout: VGPR r, lanes 0-15 -> M=r, lanes 16-31 -> M=r+8)
  // into wave-private LDS, repartition one row per lane.
  {
    float* cw = ldsC + wave * 256;
#pragma unroll
    for (int r = 0; r < 8; ++r)
      cw[(r + 8 * half) * 16 + m] = c[r];
  }
  // Same-wave LDS ops are in-order; no barrier needed before re-reading.

  if (lane < 16) {
    const float* cw = ldsC + wave * 256 + lane * 16;
    float vals[16];
#pragma unroll
    for (int i = 0; i < 16; i += 4) {
      v4f t = *(const v4f*)(cw + i);   // ds_load_b128
      vals[i] = t.x; vals[i + 1] = t.y; vals[i + 2] = t.z; vals[i + 3] = t.w;
    }
#pragma unroll
    for (int i = 0; i < 4; ++i) vals[i]     += bg[i];
#pragma unroll
    for (int i = 0; i < 4; ++i) vals[4 + i] += bp[i];
#pragma unroll
    for (int i = 0; i < 5; ++i) vals[8 + i] += ba[i];

    // softmax(ghost), softmax(pacman) — normalized (bilinear term needs it)
    float mg = fmaxf(fmaxf(vals[0], vals[1]), fmaxf(vals[2], vals[3]));
    float g0 = __expf(vals[0] - mg), g1 = __expf(vals[1] - mg);
    float g2 = __expf(vals[2] - mg), g3 = __expf(vals[3] - mg);
    float ginv = 1.0f / (g0 + g1 + g2 + g3);

    float mp = fmaxf(fmaxf(vals[4], vals[5]), fmaxf(vals[6], vals[7]));
    float p0 = __expf(vals[4] - mp), p1 = __expf(vals[5] - mp);
    float p2 = __expf(vals[6] - mp), p3 = __expf(vals[7] - mp);
    float pinv = 1.0f / (p0 + p1 + p2 + p3);

    // base softmax: its normalizer cancels in the final renormalization.
    float ma = fmaxf(fmaxf(fmaxf(vals[8], vals[9]), fmaxf(vals[10], vals[11])), vals[12]);
    float e0 = __expf(vals[8]  - ma), e1 = __expf(vals[9]  - ma);
    float e2 = __expf(vals[10] - ma), e3 = __expf(vals[11] - ma);
    float e4 = __expf(vals[12] - ma);

    // unsafe[a] = sum_{c,d} pac[c] * T[a,c,d] * ghost[d]
    float gp = ginv * pinv;
    float u0 = (p0 * g0 + p1 * g1 + p2 * g2 + p3 * g3) * gp;  // stay
    float u1 = (p0 * g1 + p2 * g3) * gp;                      // up
    float u2 = (p1 * g0 + p3 * g2) * gp;                      // down
    // left/right: no transition clause -> always safe

    float j0 = e0 * (1.0f - u0);
    float j1 = e1 * (1.0f - u1);
    float j2 = e2 * (1.0f - u2);
    float j3 = e3;
    float j4 = e4;
    float inv = 1.0f / (j0 + j1 + j2 + j3 + j4);

    float* o = out + (size_t)(rowBase + lane) * 5;
    o[0] = j0 * inv; o[1] = j1 * inv; o[2] = j2 * inv;
    o[3] = j3 * inv; o[4] = j4 * inv;
  }
}

extern "C" void kernel_launch(void* const* d_in, const int* in_sizes, int n_in,
                              void* d_out, int out_size, void* d_ws, size_t ws_size,
                              hipStream_t stream) {
  const float* x  = (const float*)d_in[0];
  const float* Wg = (const float*)d_in[1];
  const float* bg = (const float*)d_in[2];
  const float* Wp = (const float*)d_in[3];
  const float* bp = (const float*)d_in[4];
  const float* Wa = (const float*)d_in[5];
  const float* ba = (const float*)d_in[6];

  const int H = in_sizes[1] / 4;   // Wg is [H,4]
  const int B = in_sizes[0] / H;   // x  is [B,H]

  const int rowsPerBlock = 16 * WAVES_PER_BLOCK;     // 64
  const int grid = B / rowsPerBlock;                 // 16384 -> 256 blocks

  hipLaunchKernelGGL(dpl_safe_policy_kernel, dim3(grid), dim3(BLOCK), 0, stream,
                     x, Wg, bg, Wp, bp, Wa, ba, (float*)d_out, H);
}